// ModelVIGOR_3066606649384
// MI455X (gfx1250) — compile-verified
//
#include <hip/hip_runtime.h>
#include <hip/hip_bf16.h>
#include <stdint.h>

// Problem constants (match reference setup_inputs: B=8, C=64, H=256, W=512, sat_width=128)
#define B_    8
#define C_    64
#define H_    256
#define W_    512
#define SW    128
#define HALF  64
#define HW_   (H_*W_)        // 131072 == 1<<17
#define NPIX  (B_*HW_)       // 1,048,576
#define NCELL (B_*SW*SW)     // 131,072 cells -> 1 MB of u64 keys in d_ws

typedef uint32_t v4u __attribute__((ext_vector_type(4)));
typedef int      v8i __attribute__((ext_vector_type(8)));
typedef int      v4i __attribute__((ext_vector_type(4)));

// Monotonic float->uint mapping: a < b  <=>  orderF32(a) < orderF32(b)
__device__ __forceinline__ uint32_t orderF32(float f) {
  uint32_t u = __float_as_uint(f);
  return (u & 0x80000000u) ? ~u : (u | 0x80000000u);
}

// ---------------- Kernel A: reset per-cell argmin keys ----------------
__global__ void initKeys(unsigned long long* __restrict__ keys) {
  int i = blockIdx.x * blockDim.x + threadIdx.x;
  if (i < NCELL) keys[i] = ~0ull;
}

// ---------------- Kernel B: project pixels, per-cell argmin(y) via u64 atomic min ----------------
__global__ void projectKernel(const float* __restrict__ depth,
                              const float* __restrict__ mpp,
                              unsigned long long* __restrict__ keys) {
  int n = blockIdx.x * blockDim.x + threadIdx.x;
  if (n >= NPIX) return;
  int b  = n >> 17;            // / (H*W)
  int hw = n & (HW_ - 1);
  int h  = hw >> 9;            // / W (512)
  int w  = hw & (W_ - 1);

  const float TWO_PI = 6.2831853071795864f;
  const float PI     = 3.1415926535897932f;
  float theta = (float)w * (TWO_PI / (float)(W_ - 1));  // linspace(0, 2pi, W)
  float phi   = (float)h * (PI     / (float)(H_ - 1));  // linspace(0,  pi, H)
  float sp, cp, st, ct;
  __sincosf(phi,   &sp, &cp);
  __sincosf(theta, &st, &ct);

  float d = depth[n];
  float m = mpp[b];
  // xyz @ R.T with R = [[0,0,1],[0,1,0],[-1,0,0]] folds to:
  // x' = z = -sin(phi)sin(theta), y' = -cos(phi), z' = -x = -sin(phi)cos(theta)
  float xr = -sp * st;
  float yr = -cp;
  float zr = -sp * ct;

  float X = truncf(xr * d / m);
  float Z = truncf(zr * d / m);
  if (X >= -(float)HALF && X <= (float)(HALF - 1) &&
      Z >= -(float)HALF && Z <= (float)(HALF - 1)) {
    int xi = (int)X + HALF;
    int zi = (int)Z + HALF;
    int cell = (b << 14) + (xi << 7) + zi;
    float y = yr * d;
    // argmin(y); ties -> largest pixel index (matches reference lexsort keep-last)
    unsigned long long key =
        ((unsigned long long)orderF32(y) << 32) | (uint32_t)(~(uint32_t)n);
    atomicMin(&keys[cell], key);   // global_atomic_min_u64
  }
}

// ---------------- Kernel C: gather winning feature vectors (TDM-staged keys) ----------------
// One block per (b, x) satellite row: DMA the row's 128 u64 keys into LDS with the
// Tensor Data Mover, then 64 channels x 128 z-cells of coalesced output writes.
__global__ void gatherKernel(const float* __restrict__ img,
                             const unsigned long long* __restrict__ keys,
                             float* __restrict__ out) {
  __shared__ unsigned long long sKeys[SW];

  int bx = blockIdx.x;        // b*128 + x
  int b  = bx >> 7;
  int x  = bx & (SW - 1);

  if ((threadIdx.x >> 5) == 0) {   // TDM ignores EXEC: exactly one wave issues the DMA
    uint64_t gaddr  = (uint64_t)(uintptr_t)(keys + ((uint64_t)bx << 7));
    uint32_t ldsOff = (uint32_t)(uintptr_t)(void*)&sKeys[0];  // low 32 bits = LDS byte offset

    // D# group 0 (ISA 8.3): count=1 valid descriptor | lds_addr | global_addr[56:0] | type=2
    v4u g0;
    g0.x = 1u;
    g0.y = ldsOff;
    g0.z = (uint32_t)gaddr;
    g0.w = (uint32_t)((gaddr >> 32) & 0x01FFFFFFu) | 0x80000000u;  // addr[56:32] | type=2<<30

    // D# group 1 (ISA 8.4): data_size=3 (8B); tensor 128x1; tile 128x1; dim0 stride 128
    v8i g1;
    g1[0] = 0x00030000;          // wg_mask=0, data_size=8B
    g1[1] = (int)(128u << 16);   // tensor_dim0 = 128 (elements)
    g1[2] = (int)(1u   << 16);   // tensor_dim1 = 1
    g1[3] = (int)(128u << 16);   // tile_dim0   = 128
    g1[4] = 1;                   // tile_dim1   = 1
    g1[5] = 128;                 // tensor_dim0_stride = 128
    g1[6] = 0;
    g1[7] = 0;

    v4i g2 = {0, 0, 0, 0};
    v4i g3 = {0, 0, 0, 0};
    v8i g4 = {0, 0, 0, 0, 0, 0, 0, 0};   // 6-arg toolchain form: extra group before cpol
    __builtin_amdgcn_tensor_load_to_lds(g0, g1, g2, g3, g4, 0);  // tensor_load_to_lds
    __builtin_amdgcn_s_wait_tensorcnt(0);                        // s_wait_tensorcnt 0
  }
  __syncthreads();

  const float* imgB = img + (((uint64_t)b * C_) << 17);   // img[b][c][hw], c-stride = 1<<17
  const uint64_t outBase = (uint64_t)b * (C_ * SW * SW) + (uint64_t)x * SW;

  for (int e = threadIdx.x; e < C_ * SW; e += blockDim.x) {
    int c = e >> 7;          // channel
    int z = e & (SW - 1);    // z-cell (consecutive threads -> coalesced stores)
    unsigned long long key = sKeys[z];
    float v = 0.0f;
    if (key != ~0ull) {
      uint32_t n = ~(uint32_t)key;     // winning pixel flat index
      int hw = (int)(n & (HW_ - 1));
      const float* p = &imgB[(((uint64_t)c) << 17) + hw];
      v = *p;
      if (c + 2 < C_)                  // next channel this thread touches: +2*2^17 elems
        __builtin_prefetch((const void*)(p + (2u << 17)), 0, 0);  // global_prefetch_b8
    }
    out[outBase + (uint64_t)c * (SW * SW) + z] = v;
  }
}

extern "C" void kernel_launch(void* const* d_in, const int* in_sizes, int n_in,
                              void* d_out, int out_size, void* d_ws, size_t ws_size,
                              hipStream_t stream) {
  const float* img   = (const float*)d_in[0];  // [8,64,256,512] f32
  const float* depth = (const float*)d_in[1];  // [8,1,256,512]  f32
  const float* mpp   = (const float*)d_in[2];  // [8]            f32
  (void)d_in; (void)in_sizes; (void)n_in;      // d_in[3] = sat_width (128, compile-time)
  float* out = (float*)d_out;                  // [8,64,128,128] f32
  unsigned long long* keys = (unsigned long long*)d_ws;  // 1 MB scratch
  (void)out_size; (void)ws_size;

  initKeys   <<<(NCELL + 255) / 256, 256, 0, stream>>>(keys);
  projectKernel<<<(NPIX + 255) / 256, 256, 0, stream>>>(depth, mpp, keys);
  gatherKernel <<<B_ * SW,           256, 0, stream>>>(img, keys, out);
}